// UnsupLoss_2181843386910
// MI455X (gfx1250) — compile-verified
//
#include <hip/hip_runtime.h>

#define B_ 4
#define N_ 2048
#define M_ 2048
#define TILE 16
#define BLOCK_ROWS 128   // 8 waves * 16 rows per block
#define NT 256

typedef __attribute__((ext_vector_type(2))) float v2f;
typedef __attribute__((ext_vector_type(8))) float v8f;

// workspace layout (float/uint elements)
#define OFF_CM    0                       // [B*N] colmin of chamfer d (uint bits, atomic)
#define OFF_C1    (B_*N_)                 // [B*N] c1 colmin (uint bits, atomic)
#define OFF_D1    (2*B_*N_)               // [B*N] dist1 rowmin
#define OFF_MIND  (3*B_*N_)               // [B*M] min_dists^2 (y0 vs x_t0)
#define OFF_MIDX  (3*B_*N_ + B_*M_)       // [B*M] argmin indices (uint)
#define OFF_D2B   (3*B_*N_ + 2*B_*M_)     // [B*M] rowmin (y0 vs x_t1)
#define OFF_ACC   (3*B_*N_ + 3*B_*M_)     // [4] dyn, stat, dist_num, dist_den
#define OFF_MASK  (OFF_ACC + 4)           // [B] mask_chamfer

__global__ __launch_bounds__(NT)
void k_init(unsigned* __restrict__ wsu) {
  int i = blockIdx.x * blockDim.x + threadIdx.x;
  if (i < 2 * B_ * N_) wsu[i] = 0x7F800000u;     // +inf bits for the two colmin arrays
  if (i < 4 + B_)      wsu[OFF_ACC + i] = 0u;    // accumulators + mask slots
}

// One wave handles 16 rows; block handles 128 rows of batch blockIdx.y.
// D = A x B + C with A row = [ax,ay,az,a2], B col = [-2bx,-2by,-2bz,1], C = b2[n]:
// D[m,n] = a2 + b2 - 2 a.b  (full pairwise squared distance per WMMA tile).
template<bool ADD_M, bool ARGMIN, bool COLMIN>
__global__ __launch_bounds__(NT)
void k_pair(const float* __restrict__ Apts, int a_bstride,
            const float* __restrict__ Mh,
            const float* __restrict__ Bpts, int b_bstride,
            float* __restrict__ rowout,
            unsigned* __restrict__ idxout,
            unsigned* __restrict__ colmin_u)
{
  __shared__ float4   sB[N_];     // {-2x,-2y,-2z,b2}
  __shared__ unsigned sCol[N_];
  const int b   = blockIdx.y;
  const int tid = threadIdx.x;

  const float* Bb = Bpts + b * b_bstride;
  for (int n = tid; n < N_; n += NT) {
    float bx = Bb[n*3+0], by = Bb[n*3+1], bz = Bb[n*3+2];
    sB[n] = make_float4(-2.f*bx, -2.f*by, -2.f*bz, bx*bx + by*by + bz*bz);
    if (COLMIN) sCol[n] = 0x7F800000u;
  }
  __syncthreads();

  const int lane = tid & 31;
  const int wave = tid >> 5;
  const int half = lane >> 4;       // A/B K-split across wave halves
  const int mr   = lane & 15;
  const int row  = blockIdx.x * BLOCK_ROWS + wave * TILE + mr;

  const float* Ab = Apts + b * a_bstride;
  float ax = Ab[row*3+0], ay = Ab[row*3+1], az = Ab[row*3+2];
  if (ADD_M) {
    const float* Mb = Mh + b * (N_*3);
    ax += Mb[row*3+0]; ay += Mb[row*3+1]; az += Mb[row*3+2];
  }
  const float a2 = ax*ax + ay*ay + az*az;
  v2f a;                       // 16x4 f32 A: VGPR0 = K0|K2, VGPR1 = K1|K3
  a.x = half ? az : ax;
  a.y = half ? a2 : ay;

  v8f rm; unsigned ri[8];
  #pragma unroll
  for (int j = 0; j < 8; ++j) { rm[j] = 3.0e38f; ri[j] = 0u; }

  for (int c0 = 0; c0 < N_; c0 += TILE) {
    float4 q = sB[c0 + mr];
    v2f bb;                    // 4x16 f32 B: VGPR0 = K0|K2 rows, VGPR1 = K1|K3 rows
    bb.x = half ? q.z : q.x;
    bb.y = half ? 1.0f : q.y;
    v8f c;
    #pragma unroll
    for (int j = 0; j < 8; ++j) c[j] = q.w;   // C[m,n] = b2[n], n = lane&15

    v8f d = __builtin_amdgcn_wmma_f32_16x16x4_f32(
        false, a, false, bb, (short)0, c, false, false);

    if (COLMIN) {
      float cv = d[0];
      #pragma unroll
      for (int j = 1; j < 8; ++j) cv = fminf(cv, d[j]);   // min over 8 rows in-lane
      cv = fminf(cv, __shfl_xor(cv, 16, 32));             // combine wave halves
      if (half == 0)
        atomicMin(&sCol[c0 + mr], __float_as_uint(fmaxf(cv, 0.f)));
    }
    const unsigned ncol = (unsigned)(c0 + mr);
    #pragma unroll
    for (int j = 0; j < 8; ++j) {
      if (ARGMIN) {
        bool lt = d[j] < rm[j];
        rm[j] = lt ? d[j] : rm[j];
        ri[j] = lt ? ncol : ri[j];
      } else {
        rm[j] = fminf(rm[j], d[j]);
      }
    }
  }

  // row-min reduction across the 16 column lanes of each half (rows j+8*half)
  #pragma unroll
  for (int off = 1; off <= 8; off <<= 1) {
    #pragma unroll
    for (int j = 0; j < 8; ++j) {
      float ov = __shfl_xor(rm[j], off, 32);
      if (ARGMIN) {
        unsigned oi = (unsigned)__shfl_xor((int)ri[j], off, 32);
        if (ov < rm[j] || (ov == rm[j] && oi < ri[j])) { rm[j] = ov; ri[j] = oi; }
      } else {
        rm[j] = fminf(rm[j], ov);
      }
    }
  }

  if (mr == 0) {
    const int rbase = blockIdx.x * BLOCK_ROWS + wave * TILE + 8 * half;
    #pragma unroll
    for (int j = 0; j < 8; ++j) {
      rowout[b * N_ + rbase + j] = fmaxf(rm[j], 0.f);
      if (ARGMIN) idxout[b * N_ + rbase + j] = ri[j];
    }
  }

  if (COLMIN) {
    __syncthreads();
    for (int n = tid; n < N_; n += NT)
      atomicMin(&colmin_u[b * N_ + n], sCol[n]);
  }
}

// per-batch: chamfer sums -> mask_chamfer; c1 thresholded sums -> loss_dist accum
__global__ __launch_bounds__(NT)
void k_f1(float* __restrict__ ws) {
  __shared__ float4 red[NT];
  const int b = blockIdx.x, tid = threadIdx.x;
  const unsigned* wsu = (const unsigned*)ws;
  float s1 = 0.f, s2 = 0.f, num = 0.f, den = 0.f;
  for (int n = tid; n < N_; n += NT) {
    s1 += ws[OFF_D1 + b*N_ + n];
    s2 += __uint_as_float(wsu[OFF_CM + b*N_ + n]);
    float c = __uint_as_float(wsu[OFF_C1 + b*N_ + n]);
    if (c < 0.1f) { num += c; den += 1.f; }
  }
  red[tid] = make_float4(s1, s2, num, den);
  __syncthreads();
  for (int s = NT/2; s > 0; s >>= 1) {
    if (tid < s) {
      red[tid].x += red[tid+s].x; red[tid].y += red[tid+s].y;
      red[tid].z += red[tid+s].z; red[tid].w += red[tid+s].w;
    }
    __syncthreads();
  }
  if (tid == 0) {
    float lc = red[0].x / (float)N_ + red[0].y / (float)N_;
    ws[OFF_MASK + b] = (lc < 0.1f) ? 1.f : 0.f;
    atomicAdd(ws + OFF_ACC + 2, red[0].z);
    atomicAdd(ws + OFF_ACC + 3, red[0].w);
  }
}

// per-(b,m): dynamic/static loss contributions
__global__ __launch_bounds__(NT)
void k_f2(float* __restrict__ ws, const float* __restrict__ y0,
          const float* __restrict__ y1, const float* __restrict__ mh) {
  __shared__ float2 red[NT];
  const int b = blockIdx.y;
  const int m = blockIdx.x * NT + threadIdx.x;
  const unsigned* wsu = (const unsigned*)ws;
  float md     = ws[OFF_MIND + b*M_ + m];                 // squared min over x_t0
  unsigned idx = wsu[OFF_MIDX + b*M_ + m];
  float d2     = fminf(md, ws[OFF_D2B + b*M_ + m]);       // squared min over x_cat
  float mdist  = (md < 0.0025f) ? 1.f : 0.f;              // sqrt(md) < 0.05
  float mneg   = (d2 > 0.04f)   ? 1.f : 0.f;              // sqrt(d2) > 0.2
  float mc     = ws[OFF_MASK + b];
  const float* p0 = y0 + ((size_t)b*M_ + m)*3;
  const float* p1 = y1 + ((size_t)b*M_ + m)*3;
  const float* nm = mh + ((size_t)b*N_ + idx)*3;
  float dyn = 0.f, sta = 0.f;
  #pragma unroll
  for (int k = 0; k < 3; ++k) {
    float diff = p1[k] - p0[k];
    float pr = diff * mc * mdist;
    float tg = nm[k] * mc * mdist;
    float e = pr - tg; dyn += e * e;
    float s = diff * mneg; sta += s * s;
  }
  red[threadIdx.x] = make_float2(dyn, sta);
  __syncthreads();
  for (int s = NT/2; s > 0; s >>= 1) {
    if ((int)threadIdx.x < s) {
      red[threadIdx.x].x += red[threadIdx.x+s].x;
      red[threadIdx.x].y += red[threadIdx.x+s].y;
    }
    __syncthreads();
  }
  if (threadIdx.x == 0) {
    atomicAdd(ws + OFF_ACC + 0, red[0].x);
    atomicAdd(ws + OFF_ACC + 1, red[0].y);
  }
}

__global__ void k_f3(const float* __restrict__ ws, float* __restrict__ out) {
  out[0] = ws[OFF_ACC + 0] / (float)(B_ * M_ * 3);
  out[1] = ws[OFF_ACC + 1] / (float)(B_ * M_ * 3);
  out[2] = ws[OFF_ACC + 2] / fmaxf(ws[OFF_ACC + 3], 1.f);
}

extern "C" void kernel_launch(void* const* d_in, const int* in_sizes, int n_in,
                              void* d_out, int out_size, void* d_ws, size_t ws_size,
                              hipStream_t stream) {
  (void)in_sizes; (void)n_in; (void)out_size; (void)ws_size;
  const float* x  = (const float*)d_in[0];   // (B,2,N,3)
  const float* mh = (const float*)d_in[1];   // (B,1,N,3)
  const float* y0 = (const float*)d_in[2];   // (B,1,M,3)
  const float* y1 = (const float*)d_in[3];   // (B,1,M,3)
  float*    out  = (float*)d_out;
  float*    ws   = (float*)d_ws;
  unsigned* wsu  = (unsigned*)d_ws;

  k_init<<<(2*B_*N_ + NT - 1)/NT, NT, 0, stream>>>(wsu);

  dim3 grid(N_ / BLOCK_ROWS, B_);   // 16 x 4

  // P1: chamfer — rows = x_t0 + m_hat, cols = x_t1
  k_pair<true,  false, true ><<<grid, NT, 0, stream>>>(
      x,          2*N_*3, mh, x + N_*3, 2*N_*3,
      ws + OFF_D1, nullptr, wsu + OFF_CM);

  // P2: rows = y0, cols = x_t0 — rowmin+argmin (min_dists, indices), colmin (c1)
  k_pair<false, true,  true ><<<grid, NT, 0, stream>>>(
      y0,         M_*3,  nullptr, x,        2*N_*3,
      ws + OFF_MIND, wsu + OFF_MIDX, wsu + OFF_C1);

  // P3: rows = y0, cols = x_t1 — rowmin only (second half of x_cat)
  k_pair<false, false, false><<<grid, NT, 0, stream>>>(
      y0,         M_*3,  nullptr, x + N_*3, 2*N_*3,
      ws + OFF_D2B, nullptr, nullptr);

  k_f1<<<B_, NT, 0, stream>>>(ws);
  dim3 g2(M_ / NT, B_);
  k_f2<<<g2, NT, 0, stream>>>(ws, y0, y1, mh);
  k_f3<<<1, 1, 0, stream>>>(ws, out);
}